// HyperGraphSAGE_88055419503320
// MI455X (gfx1250) — compile-verified
//
#include <hip/hip_runtime.h>

typedef __attribute__((ext_vector_type(2))) float v2f;
typedef __attribute__((ext_vector_type(8))) float v8f;

#define F 128  // in_f == out_f

// ---------------------------------------------------------------------------
// GEMM: Hp = H * W using V_WMMA_F32_16X16X4_F32 (fp32, matches reference).
// One wave computes a full 16x128 output strip: A-fragment loaded once per
// k-step and reused across all 8 n-tiles (8x less H traffic than tile-per-wave).
// ---------------------------------------------------------------------------
__global__ __launch_bounds__(256)
void gat_gemm_wmma(const float* __restrict__ H, const float* __restrict__ W,
                   float* __restrict__ Hp, int N) {
  const int lane = threadIdx.x & 31;
  const int wave = (int)((blockIdx.x * blockDim.x + threadIdx.x) >> 5);
  const int mTile = wave;                       // one 16-row strip per wave
  if (mTile * 16 >= N) return;

  // A-frag (16x4 f32): lanes 0-15 -> K,K+1 ; lanes 16-31 -> K+2,K+3 (M = lane&15)
  const int mr = lane & 15;
  const int kk = (lane >> 4) * 2;
  const float* __restrict__ Arow = H + (size_t)(mTile * 16 + mr) * F;
  // B-frag (4x16 f32): N = lane&15 within each n-tile, same K split
  const float* __restrict__ Bbase = W + (lane & 15);

  v8f acc[8];
#pragma unroll
  for (int nt = 0; nt < 8; ++nt) acc[nt] = (v8f){};

#pragma unroll 2
  for (int k = 0; k < F; k += 4) {
    v2f a = *(const v2f*)(Arow + k + kk);        // contiguous pair, 8B aligned
    const float* __restrict__ Brow0 = Bbase + (size_t)(k + kk) * F;
    const float* __restrict__ Brow1 = Brow0 + F;
#pragma unroll
    for (int nt = 0; nt < 8; ++nt) {
      v2f b;
      b.x = Brow0[nt * 16];
      b.y = Brow1[nt * 16];
      acc[nt] = __builtin_amdgcn_wmma_f32_16x16x4_f32(false, a, false, b,
                                                      (short)0, acc[nt],
                                                      false, false);
    }
  }

  // C/D layout: lane L, VGPR r -> (M = (L>=16 ? 8 : 0) + r, N = L&15)
  const int mBase = (lane >> 4) * 8;
  float* __restrict__ outBase =
      Hp + (size_t)(mTile * 16 + mBase) * F + (lane & 15);
#pragma unroll
  for (int nt = 0; nt < 8; ++nt) {
#pragma unroll
    for (int r = 0; r < 8; ++r)
      outBase[(size_t)r * F + nt * 16] = acc[nt][r];
  }
}

// ---------------------------------------------------------------------------
// Fused edge pass: score -> leaky_relu -> exp -> accumulate numerator+denominator
// One edge per wave32 iteration; lane holds 4 of 128 features.
// Softmax denominator divided out in the finalize pass (per-row scalar), so
// edges are visited exactly once.
// ---------------------------------------------------------------------------
__global__ __launch_bounds__(256)
void gat_edge(const float* __restrict__ Hp, const int* __restrict__ row,
              const int* __restrict__ col, float* __restrict__ denom,
              float* __restrict__ outAcc, int E) {
  const int lane = threadIdx.x & 31;
  const int wave = (int)((blockIdx.x * blockDim.x + threadIdx.x) >> 5);
  const int nWaves = (int)((gridDim.x * blockDim.x) >> 5);

  for (int e = wave; e < E; e += nWaves) {
    const int r = row[e];
    const int c = col[e];

    // prefetch next edge's rows (L2-resident Hp) -> global_prefetch_b8
    const int e2 = e + nWaves;
    if (e2 < E) {
      const int rn = row[e2];
      const int cn = col[e2];
      __builtin_prefetch(Hp + (size_t)rn * F + lane * 4, 0, 3);
      __builtin_prefetch(Hp + (size_t)cn * F + lane * 4, 0, 3);
    }

    const float4 hr = *(const float4*)(Hp + (size_t)r * F + lane * 4);
    const float4 hc = *(const float4*)(Hp + (size_t)c * F + lane * 4);

    float p = hr.x * hc.x + hr.y * hc.y + hr.z * hc.z + hr.w * hc.w;
#pragma unroll
    for (int off = 16; off >= 1; off >>= 1) p += __shfl_xor(p, off, 32);

    const float s = fmaxf(p, 0.2f * p);   // leaky_relu, slope 0.2
    const float ex = __expf(s);

    if (lane == 0) atomicAdd(denom + r, ex);

    float* o = outAcc + (size_t)r * F + lane * 4;
    atomicAdd(o + 0, ex * hc.x);
    atomicAdd(o + 1, ex * hc.y);
    atomicAdd(o + 2, ex * hc.z);
    atomicAdd(o + 3, ex * hc.w);
  }
}

// ---------------------------------------------------------------------------
__global__ void zero_kernel(float* __restrict__ p, size_t n) {
  size_t i = (size_t)blockIdx.x * blockDim.x + threadIdx.x;
  const size_t stride = (size_t)gridDim.x * blockDim.x;
  for (; i < n; i += stride) p[i] = 0.0f;
}

__global__ void gat_finalize(float* __restrict__ out, const float* __restrict__ denom,
                             const float* __restrict__ bias, size_t n) {
  size_t i = (size_t)blockIdx.x * blockDim.x + threadIdx.x;
  const size_t stride = (size_t)gridDim.x * blockDim.x;
  for (; i < n; i += stride) {
    const size_t node = i >> 7;          // /128
    const int cc = (int)(i & (F - 1));
    out[i] = out[i] / (denom[node] + 1e-10f) + bias[cc];
  }
}

// ---------------------------------------------------------------------------
extern "C" void kernel_launch(void* const* d_in, const int* in_sizes, int n_in,
                              void* d_out, int out_size, void* d_ws, size_t ws_size,
                              hipStream_t stream) {
  const float* H    = (const float*)d_in[0];
  const float* W    = (const float*)d_in[1];
  const float* bias = (const float*)d_in[2];
  const int*   row  = (const int*)d_in[3];
  const int*   col  = (const int*)d_in[4];

  const int N = in_sizes[0] / F;   // 100000 nodes
  const int E = in_sizes[3];       // 1.6M edges
  float* out = (float*)d_out;

  // workspace: Hp [N*F] f32, then denom [N] f32
  float* Hp    = (float*)d_ws;
  float* denom = (float*)((char*)d_ws + (size_t)N * F * sizeof(float));

  // zero accumulators (harness poisons buffers)
  zero_kernel<<<512, 256, 0, stream>>>(denom, (size_t)N);
  zero_kernel<<<4096, 256, 0, stream>>>(out, (size_t)N * F);

  // GEMM: one wave per 16-row strip; 8 waves per block
  const int strips = (N + 15) / 16;                 // 6250
  gat_gemm_wmma<<<(strips + 7) / 8, 256, 0, stream>>>(H, W, Hp, N);

  // fused edge pass
  gat_edge<<<4096, 256, 0, stream>>>(Hp, row, col, denom, out, E);

  // normalize + bias
  gat_finalize<<<4096, 256, 0, stream>>>(out, denom, bias, (size_t)N * F);
}